// TGNNNode_24472723652617
// MI455X (gfx1250) — compile-verified
//
#include <hip/hip_runtime.h>
#include <hip/hip_bf16.h>

typedef __attribute__((ext_vector_type(2))) float v2f;
typedef __attribute__((ext_vector_type(8))) float v8f;

#define FX 64
#define FU 128
#define FH 192     // FX + FU
#define FOUT 128
#define LDSROW 196 // padded LDS row stride (196 % 64 = 4 -> conflict-free b64 frags)

// ---------------------------------------------------------------------------
// Kernel 1: build h[N,192] in workspace.
//   cols 0..63   -> 0 (scatter-add target)
//   cols 64..191 -> u[batch[i]]  (fused concat)
// ---------------------------------------------------------------------------
__global__ void tgnn_init_h(const float* __restrict__ u,
                            const int* __restrict__ batch,
                            float* __restrict__ h, int N) {
    int idx = blockIdx.x * blockDim.x + threadIdx.x;
    int total = N * FH;
    if (idx >= total) return;
    int i = idx / FH;
    int j = idx - i * FH;
    float v = 0.0f;
    if (j >= FX) {
        v = u[(size_t)batch[i] * FU + (j - FX)];
    }
    h[idx] = v;
}

// ---------------------------------------------------------------------------
// Kernel 2: edge scatter-add. 16 threads per edge; thread t does a coalesced
// float4 read of x[src] and 4 f32 atomicAdds into h[dest][t*4..].
// x (25.6MB) and h (76.8MB) are L2-resident on MI455X (192MB L2); the
// once-streamed edge arrays are loaded non-temporally so they don't evict
// that working set.
// ---------------------------------------------------------------------------
__global__ void tgnn_edge_scatter(const float4* __restrict__ x4,
                                  const int* __restrict__ src,
                                  const int* __restrict__ dst,
                                  const float* __restrict__ attr,
                                  float* __restrict__ h, int E) {
    int gid = blockIdx.x * blockDim.x + threadIdx.x;
    int e = gid >> 4;
    int t = gid & 15;
    if (e >= E) return;
    int s   = __builtin_nontemporal_load(src + e);
    int d   = __builtin_nontemporal_load(dst + e);
    float a = __builtin_nontemporal_load(attr + e);
    float4 v = x4[(size_t)s * (FX / 4) + t];
    float* hp = h + (size_t)d * FH + t * 4;
    atomicAdd(hp + 0, a * v.x);
    atomicAdd(hp + 1, a * v.y);
    atomicAdd(hp + 2, a * v.z);
    atomicAdd(hp + 3, a * v.w);
}

// ---------------------------------------------------------------------------
// Kernel 3: [N,192] x [192,256] GEMM via V_WMMA_F32_16X16X4_F32.
// One 512-thread block (16 waves) per 16-row tile; each wave owns one 16-col
// tile (cols 0..127 = K, 128..255 = Q). The 16x192 A-tile is staged once per
// block into LDS (coalesced float2 loads), then each wave reads A fragments
// via ds_load_b64 (bank-conflict-free with the 196-float row pad).
//
// f32 WMMA lane layouts (ISA 7.12.2):
//   A 16x4 : m=lane&15, half=lane>>4; VGPR0 = K=2*half, VGPR1 = K=2*half+1
//   B 4x16 : n=lane&15;               VGPR0 = row 2*half, VGPR1 = row 2*half+1
//   C 16x16: VGPR v -> row (v + 8*half), col (lane&15)
// ---------------------------------------------------------------------------
__global__ void __launch_bounds__(512)
tgnn_gemm_wmma(const float* __restrict__ h,
               const float* __restrict__ W_K,
               const float* __restrict__ b_K,
               const float* __restrict__ W_Q,
               const float* __restrict__ b_Q,
               float* __restrict__ out, int N) {
    __shared__ float hs[16 * LDSROW];

    const int tid  = threadIdx.x;
    const int lane = tid & 31;
    const int wave = tid >> 5;              // 0..15 -> column tile
    const int row0 = blockIdx.x * 16;

    // --- cooperative A-tile stage: 16 rows x 192 cols as 1536 float2 ---
    for (int k = tid; k < 16 * (FH / 2); k += 512) {
        int r  = k / (FH / 2);
        int c2 = k - r * (FH / 2);
        int rr = row0 + r;
        if (rr >= N) rr = N - 1;            // clamp for tail tiles
        float2 v = *(const float2*)(h + (size_t)rr * FH + c2 * 2);
        *(float2*)(&hs[r * LDSROW + c2 * 2]) = v;
    }
    __syncthreads();

    const int col0 = wave * 16;
    const int m    = lane & 15;
    const int half = lane >> 4;             // 0 or 1
    const int koff = half * 2;

    const bool   isK  = (col0 < FOUT);
    const float* W    = isK ? W_K : W_Q;
    const float* bia  = isK ? b_K : b_Q;
    const int    wcol = isK ? col0 : (col0 - FOUT);

    const float* arow = &hs[m * LDSROW];

    v8f acc = {};
    #pragma unroll 4
    for (int kk = 0; kk < FH; kk += 4) {
        float2 av = *(const float2*)(arow + kk + koff);   // ds_load_b64
        v2f a; a.x = av.x; a.y = av.y;
        v2f b;
        b.x = W[(size_t)(kk + koff    ) * FOUT + wcol + m];
        b.y = W[(size_t)(kk + koff + 1) * FOUT + wcol + m];
        acc = __builtin_amdgcn_wmma_f32_16x16x4_f32(
            /*neg_a=*/false, a, /*neg_b=*/false, b,
            /*c_mod=*/(short)0, acc, /*reuse_a=*/false, /*reuse_b=*/false);
    }

    const float  bv    = bia[wcol + m];
    const size_t obase = isK ? 0 : (size_t)N * FOUT;

    if (row0 + 16 <= N) {
        // uniform fast path (always taken for N % 16 == 0): no exec juggling,
        // write-once outputs bypass cache retention via non-temporal stores
        #pragma unroll
        for (int v = 0; v < 8; ++v) {
            int r = row0 + v + half * 8;
            __builtin_nontemporal_store(acc[v] + bv,
                &out[obase + (size_t)r * FOUT + wcol + m]);
        }
    } else {
        #pragma unroll
        for (int v = 0; v < 8; ++v) {
            int r = row0 + v + half * 8;
            if (r < N) out[obase + (size_t)r * FOUT + wcol + m] = acc[v] + bv;
        }
    }
}

// ---------------------------------------------------------------------------
// Host-side launcher
// Inputs: 0:x [N,64] f32, 1:edge_index [2,E] int, 2:edge_attr [E,1] f32,
//         3:u [B,128] f32, 4:batch [N] int, 5:W_K [192,128] f32, 6:b_K [128],
//         7:W_Q [192,128] f32, 8:b_Q [128]
// Output: K flat (N*128) then Q flat (N*128), f32.
// Workspace: h [N,192] f32 = 76.8 MB.
// ---------------------------------------------------------------------------
extern "C" void kernel_launch(void* const* d_in, const int* in_sizes, int n_in,
                              void* d_out, int out_size, void* d_ws, size_t ws_size,
                              hipStream_t stream) {
    const float* x    = (const float*)d_in[0];
    const int*   ei   = (const int*)d_in[1];
    const float* attr = (const float*)d_in[2];
    const float* u    = (const float*)d_in[3];
    const int*   bat  = (const int*)d_in[4];
    const float* W_K  = (const float*)d_in[5];
    const float* b_K  = (const float*)d_in[6];
    const float* W_Q  = (const float*)d_in[7];
    const float* b_Q  = (const float*)d_in[8];
    float*       out  = (float*)d_out;

    const int N = in_sizes[0] / FX;     // 100000
    const int E = in_sizes[2];          // 3200000 (edge_attr flat count)

    const int* src = ei;                // edge_index row 0
    const int* dst = ei + E;            // edge_index row 1

    float* h = (float*)d_ws;            // [N,192]

    // 1) build h: zeros | u[batch]
    {
        int total = N * FH;
        int blocks = (total + 255) / 256;
        tgnn_init_h<<<blocks, 256, 0, stream>>>(u, bat, h, N);
    }

    // 2) scatter-add edges into h[:, 0:64]
    {
        long long total = (long long)E * 16;
        int blocks = (int)((total + 255) / 256);
        tgnn_edge_scatter<<<blocks, 256, 0, stream>>>(
            (const float4*)x, src, dst, attr, h, E);
    }

    // 3) WMMA GEMM -> K and Q (one block per row tile, 16 waves = 16 col tiles)
    {
        dim3 grid((N + 15) / 16);
        tgnn_gemm_wmma<<<grid, 512, 0, stream>>>(h, W_K, b_K, W_Q, b_Q, out, N);
    }
}